// SPDModel_71210557768066
// MI455X (gfx1250) — compile-verified
//
#include <hip/hip_runtime.h>

typedef __attribute__((ext_vector_type(2))) float v2f;
typedef __attribute__((ext_vector_type(8))) float v8f;

#define LDA0 68    // stride of square SPD buffer (up to 66x66)
#define LDA1 48    // stride of tall/skinny buffers (up to 66x48)
#define LDK  129   // chunk stride: odd -> perfect LDS bank spread
#define KC   128   // covariance K-panel width (power of two)

// Shared pool carving (floats):
//   sM0   [0,     4488)   66*68 square working buffer
//   sM1   [4488,  7656)   66*48 intermediate
//   sWS   [7656, 10824)   66*48 staged weights
//   sVS   [10824,12984)   45*48 eigenvectors
//   chunk [4488, 13002)   66*129 xc panel (cov phase only; overlaps sM1/sWS/sVS)
#define POOLSZ 13008

// ---------------------------------------------------------------------------
// Branch-free bounds-checked fetch of logical element (r,c) of an R x C
// matrix (phys. transposed if trans). Unconditional load of a clamped
// address + select -> ds_load + v_cndmask, no EXEC save/restore chains.
// ---------------------------------------------------------------------------
__device__ __forceinline__ float ldchk(const float* P, int ld, bool trans,
                                       int r, int c, int R, int C) {
  const bool ok = (r < R) && (c < C);
  int idx = trans ? (c * ld + r) : (r * ld + c);
  idx = ok ? idx : 0;
  const float v = P[idx];
  return ok ? v : 0.0f;
}

// ---------------------------------------------------------------------------
// Block-cooperative fp32 GEMM via V_WMMA_F32_16X16X4_F32 (LDS sources).
// C(MxN) = Alog(MxK) @ Blog(KxN); 16x16 tiles distributed over the 8 waves.
// A frag (16x4 f32): lanes 0-15 K=k+{0,1}; lanes 16-31 K=k+{2,3}.
// C/D: VGPR v holds rows m0+v (lanes 0-15) / m0+v+8 (lanes 16-31).
// ---------------------------------------------------------------------------
__device__ void block_gemm_wmma(const float* A, int lda, bool ta,
                                const float* B, int ldb, bool tb,
                                float* C, int ldc, int M, int N, int K) {
  const int lane = threadIdx.x & 31;
  const int wave = threadIdx.x >> 5;
  const int nw   = blockDim.x >> 5;
  const int half = lane >> 4;
  const int l    = lane & 15;
  const int MT = (M + 15) >> 4, NT = (N + 15) >> 4;
  for (int t = wave; t < MT * NT; t += nw) {
    const int m0 = (t / NT) << 4;
    const int n0 = (t % NT) << 4;
    v8f acc = {};
    for (int k = 0; k < K; k += 4) {
      const int ka = k + 2 * half;
      v2f a, b;
      a.x = ldchk(A, lda, ta, m0 + l, ka,     M, K);
      a.y = ldchk(A, lda, ta, m0 + l, ka + 1, M, K);
      b.x = ldchk(B, ldb, tb, ka,     n0 + l, K, N);
      b.y = ldchk(B, ldb, tb, ka + 1, n0 + l, K, N);
      acc = __builtin_amdgcn_wmma_f32_16x16x4_f32(
          false, a, false, b, (short)0, acc, false, false);
    }
#pragma unroll
    for (int v = 0; v < 8; ++v) {
      const int r = m0 + v + 8 * half;
      const int c = n0 + l;
      if (r < M && c < N) C[r * ldc + c] = acc[v];
    }
  }
}

// ---------------------------------------------------------------------------
// Round-robin tournament pairing: round r, pair k of m players (m even).
// ---------------------------------------------------------------------------
__device__ __forceinline__ void jac_pair(int k, int r, int m, int& p, int& q) {
  const int a = (k == 0) ? 0 : ((k - 1 + r) % (m - 1)) + 1;
  const int sB = m - 1 - k;
  const int b = ((sB - 1 + r) % (m - 1)) + 1;
  p = a < b ? a : b;
  q = a < b ? b : a;
}

// ---------------------------------------------------------------------------
// Parallel two-sided cyclic Jacobi for symmetric n x n (n <= 45).
// Phase 0: one thread per pair computes (p,q,c,s) into LDS.
// Phase 1: A = J^T A  (disjoint row pairs; wave per pair, lane per column).
// Phase 2: A = A J, V = V J (disjoint column pairs).
// No runtime integer division in the hot phases.
// ---------------------------------------------------------------------------
__device__ void jacobi_eig(float* A, int lda, float* V, int ldv, int n,
                           float* cs, float* sn, int* sP, int* sQ, int nsweep) {
  const int tid = threadIdx.x;
  const int lane = tid & 31, wave = tid >> 5, nw = blockDim.x >> 5;
  const int m = (n + 1) & ~1;
  const int npairs = m >> 1;
  for (int r = wave; r < n; r += nw)
    for (int c = lane; c < n; c += 32)
      V[r * ldv + c] = (r == c) ? 1.0f : 0.0f;
  __syncthreads();
  for (int sweep = 0; sweep < nsweep; ++sweep) {
    for (int r = 0; r < m - 1; ++r) {
      if (tid < npairs) {
        int p, q; jac_pair(tid, r, m, p, q);
        float c = 1.0f, s = 0.0f;
        if (q < n) {
          const float apq = A[p * lda + q];
          if (fabsf(apq) > 1e-20f) {
            const float app = A[p * lda + p];
            const float aqq = A[q * lda + q];
            const float th = (aqq - app) / (2.0f * apq);
            const float tt = copysignf(1.0f, th) /
                             (fabsf(th) + sqrtf(th * th + 1.0f));
            c = 1.0f / sqrtf(1.0f + tt * tt);
            s = tt * c;
          }
        }
        sP[tid] = p; sQ[tid] = q; cs[tid] = c; sn[tid] = s;
      }
      __syncthreads();
      for (int k = wave; k < npairs; k += nw) {      // rows: A = J^T A
        const int p = sP[k], q = sQ[k];
        if (q < n) {
          const float c = cs[k], s = sn[k];
          for (int j = lane; j < n; j += 32) {
            const float ap = A[p * lda + j], aq = A[q * lda + j];
            A[p * lda + j] = c * ap - s * aq;
            A[q * lda + j] = s * ap + c * aq;
          }
        }
      }
      __syncthreads();
      for (int k = wave; k < npairs; k += nw) {      // cols: A = A J, V = V J
        const int p = sP[k], q = sQ[k];
        if (q < n) {
          const float c = cs[k], s = sn[k];
          for (int j = lane; j < n; j += 32) {
            const float ap = A[j * lda + p], aq = A[j * lda + q];
            A[j * lda + p] = c * ap - s * aq;
            A[j * lda + q] = s * ap + c * aq;
            const float vp = V[j * ldv + p], vq = V[j * ldv + q];
            V[j * ldv + p] = c * vp - s * vq;
            V[j * ldv + q] = s * vp + c * vq;
          }
        }
      }
      __syncthreads();
    }
  }
}

// ---------------------------------------------------------------------------
// Single-workgroup SPDNet forward: cov -> (BiMap+ReEig)x3 -> LogEig.
// ---------------------------------------------------------------------------
__global__ __launch_bounds__(256)
void spdnet_kernel(const float* __restrict__ x,    // 66 x 625
                   const float* __restrict__ W1,   // 66 x 45
                   const float* __restrict__ W2,   // 45 x 30
                   const float* __restrict__ W3,   // 30 x 25
                   float* __restrict__ out) {      // 25 x 25
  __shared__ float pool[POOLSZ];
  __shared__ float sMean[68];
  __shared__ float sCs[24], sSn[24];
  __shared__ int   sP[24], sQ[24];
  float* const sM0 = pool;
  float* const sM1 = pool + 4488;
  float* const sWS = pool + 7656;
  float* const sVS = pool + 10824;
  float* const chk = pool + 4488;      // cov-phase xc panel (66 x KC, ld=LDK)

  const int tid = threadIdx.x, nth = blockDim.x;
  const int lane = tid & 31, wave = tid >> 5, nw = nth >> 5;
  const int half = lane >> 4, l = lane & 15;

  // Warm L2: CDNA5 global_prefetch_b8 over the whole input (165 KB).
  for (int i = tid * 64; i < 66 * 625; i += nth * 64)
    __builtin_prefetch(x + i, 0, 0);

  // Phase 0: row means (one wave per row, lane-strided reduce).
  for (int r = wave; r < 66; r += nw) {
    float s = 0.0f;
    for (int c = lane; c < 625; c += 32) s += x[r * 625 + c];
#pragma unroll
    for (int off = 16; off > 0; off >>= 1) s += __shfl_down(s, off, 32);
    if (lane == 0) sMean[r] = s * (1.0f / 625.0f);
  }
  __syncthreads();

  // Phase 1: cov(66x66) = xc @ xc^T / 624 + 1e-8*I, K-panelled through LDS.
  {
    v8f acc[4] = {{}, {}, {}, {}};
    int rA[4], rB[4];
    float mA[4], mB[4];
#pragma unroll
    for (int ti = 0; ti < 4; ++ti) {
      const int t = wave + 8 * ti;           // tile id in 5x5 grid (t<25)
      const int m0 = (t / 5) << 4, n0 = (t % 5) << 4;
      const int ra = m0 + l, rb = n0 + l;
      rA[ti] = (ra < 66) ? ra : 0;  mA[ti] = (ra < 66) ? 1.0f : 0.0f;
      rB[ti] = (rb < 66) ? rb : 0;  mB[ti] = (rb < 66) ? 1.0f : 0.0f;
    }
    for (int k0 = 0; k0 < 625; k0 += KC) {
      // Stage mean-subtracted panel, branch-free, coalesced (8448 elems).
      for (int idx = tid; idx < 66 * KC; idx += nth) {
        const int r = idx >> 7, c = idx & (KC - 1);
        const int gk = k0 + c;
        const int gks = (gk < 625) ? gk : 624;
        const float msk = (gk < 625) ? 1.0f : 0.0f;
        chk[r * LDK + c] = (x[r * 625 + gks] - sMean[r]) * msk;
      }
      __syncthreads();
#pragma unroll
      for (int ti = 0; ti < 4; ++ti) {
        if (wave + 8 * ti < 25) {
          const float* pa = chk + rA[ti] * LDK;
          const float* pb = chk + rB[ti] * LDK;
          v8f a8 = acc[ti];
#pragma unroll 4
          for (int kk = 0; kk < KC; kk += 4) {
            const int ka = kk + 2 * half;
            v2f a, b;
            a.x = pa[ka] * mA[ti];  a.y = pa[ka + 1] * mA[ti];
            b.x = pb[ka] * mB[ti];  b.y = pb[ka + 1] * mB[ti];
            a8 = __builtin_amdgcn_wmma_f32_16x16x4_f32(
                false, a, false, b, (short)0, a8, false, false);
          }
          acc[ti] = a8;
        }
      }
      __syncthreads();
    }
    const float inv = 1.0f / 624.0f;
#pragma unroll
    for (int ti = 0; ti < 4; ++ti) {
      const int t = wave + 8 * ti;
      if (t < 25) {
        const int m0 = (t / 5) << 4, n0 = (t % 5) << 4;
#pragma unroll
        for (int v = 0; v < 8; ++v) {
          const int r = m0 + v + 8 * half, c = n0 + l;
          if (r < 66 && c < 66)
            sM0[r * LDA0 + c] = acc[ti][v] * inv + ((r == c) ? 1e-8f : 0.0f);
        }
      }
    }
  }
  __syncthreads();

  // Phase 2: three BiMap + ReEig layers, then LogEig.
  const float* Ws[3] = {W1, W2, W3};
  const int niA[3] = {66, 45, 30};
  const int noA[3] = {45, 30, 25};
  int n = 66;

  for (int layer = 0; layer < 3; ++layer) {
    const int a = niA[layer], b = noA[layer];
    for (int r = wave; r < a; r += nw)               // stage W (a x b)
      for (int c = lane; c < b; c += 32)
        sWS[r * LDA1 + c] = Ws[layer][r * b + c];
    __syncthreads();
    block_gemm_wmma(sM0, LDA0, false, sWS, LDA1, false, sM1, LDA1, a, b, a);
    __syncthreads();                                  // T = X @ W
    block_gemm_wmma(sWS, LDA1, true, sM1, LDA1, false, sM0, LDA0, b, b, a);
    __syncthreads();                                  // S = W^T @ T
    jacobi_eig(sM0, LDA0, sVS, LDA1, b, sCs, sSn, sP, sQ, 12);
    for (int r = wave; r < b; r += nw)                // tmp = V * max(d,eps)
      for (int c = lane; c < b; c += 32)
        sM1[r * LDA1 + c] = sVS[r * LDA1 + c] * fmaxf(sM0[c * LDA0 + c], 1e-4f);
    __syncthreads();
    block_gemm_wmma(sM1, LDA1, false, sVS, LDA1, true, sM0, LDA0, b, b, b);
    __syncthreads();                                  // X = tmp @ V^T
    n = b;
  }

  jacobi_eig(sM0, LDA0, sVS, LDA1, n, sCs, sSn, sP, sQ, 12);
  for (int r = wave; r < n; r += nw)                  // tmp = V * log(d)
    for (int c = lane; c < n; c += 32)
      sM1[r * LDA1 + c] =
          sVS[r * LDA1 + c] * logf(fmaxf(sM0[c * LDA0 + c], 1e-30f));
  __syncthreads();
  block_gemm_wmma(sM1, LDA1, false, sVS, LDA1, true, sM0, LDA0, n, n, n);
  __syncthreads();

  for (int r = wave; r < n; r += nw)
    for (int c = lane; c < n; c += 32)
      out[r * n + c] = sM0[r * LDA0 + c];
}

extern "C" void kernel_launch(void* const* d_in, const int* in_sizes, int n_in,
                              void* d_out, int out_size, void* d_ws, size_t ws_size,
                              hipStream_t stream) {
  (void)in_sizes; (void)n_in; (void)out_size; (void)d_ws; (void)ws_size;
  const float* x  = (const float*)d_in[0];   // spd_A, 66 x 625
  const float* W1 = (const float*)d_in[1];   // 66 x 45
  const float* W2 = (const float*)d_in[2];   // 45 x 30
  const float* W3 = (const float*)d_in[3];   // 30 x 25
  float* out = (float*)d_out;                // 625 floats
  spdnet_kernel<<<dim3(1), dim3(256), 0, stream>>>(x, W1, W2, W3, out);
}